// TokenEmbedding_10780367913070
// MI455X (gfx1250) — compile-verified
//
#include <hip/hip_runtime.h>
#include <hip/hip_bf16.h>

#define HIDDEN     128
#define NUM_BASES  8
#define NUM_REL    48
#define TPB        8          // edge tiles (of 16 edges) per block
#define EPB        (16 * TPB) // 128 edges per block

typedef float v2f __attribute__((ext_vector_type(2)));
typedef float v8f __attribute__((ext_vector_type(8)));

// Async DMA: global -> LDS, 16 bytes per lane, tracked by ASYNCcnt.
__device__ __forceinline__ void async_copy_b128(unsigned lds_off, const void* gaddr) {
    asm volatile("global_load_async_to_lds_b128 %0, %1, off"
                 :: "v"(lds_off), "v"(gaddr) : "memory");
}
__device__ __forceinline__ void async_wait0() {
    asm volatile("s_wait_asynccnt 0" ::: "memory");
}
__device__ __forceinline__ unsigned lds_offset_of(const void* p) {
    // generic pointers to LDS carry the LDS byte offset in their low 32 bits
    return (unsigned)(unsigned long long)p;
}

// ---------------------------------------------------------------------------
// 1) zero the (rel, dst) degree histogram
// ---------------------------------------------------------------------------
__global__ void k_zero_u32(unsigned* __restrict__ p, int n) {
    int i = blockIdx.x * blockDim.x + threadIdx.x;
    if (i < n) p[i] = 0u;
}

// ---------------------------------------------------------------------------
// 2) histogram: cnt[rel*N + dst] += 1 per edge
// ---------------------------------------------------------------------------
__global__ void k_hist(const int* __restrict__ et, const int* __restrict__ dst,
                       unsigned* __restrict__ cnt, int E, int N) {
    int e = blockIdx.x * blockDim.x + threadIdx.x;
    if (e < E) {
        atomicAdd(&cnt[(size_t)et[e] * N + dst[e]], 1u);
    }
}

// ---------------------------------------------------------------------------
// 3) pack basis/root into WMMA B-operand pair layout:
//    P[(b*64 + ks*2 + half)*128 + col] = { B[4ks+2h][col], B[4ks+2h+1][col] }
//    -> hot loop issues one global_load_b64 per WMMA.
// ---------------------------------------------------------------------------
__global__ void k_pack_basis(const float* __restrict__ basis, float2* __restrict__ P) {
    int i = blockIdx.x * blockDim.x + threadIdx.x;   // 8*32*2*128 = 65536
    if (i < NUM_BASES * 32 * 2 * HIDDEN) {
        const int col = i & 127;
        const int h   = (i >> 7) & 1;
        const int ks  = (i >> 8) & 31;
        const int b   = i >> 13;
        const int k0  = ks * 4 + h * 2;
        P[i] = make_float2(basis[((size_t)b * HIDDEN + k0) * HIDDEN + col],
                           basis[((size_t)b * HIDDEN + k0 + 1) * HIDDEN + col]);
    }
}

__global__ void k_pack_root(const float* __restrict__ root, float2* __restrict__ P) {
    int i = blockIdx.x * blockDim.x + threadIdx.x;   // 32*2*128 = 8192
    if (i < 32 * 2 * HIDDEN) {
        const int col = i & 127;
        const int h   = (i >> 7) & 1;
        const int ks  = i >> 8;
        const int k0  = ks * 4 + h * 2;
        P[i] = make_float2(root[(size_t)k0 * HIDDEN + col],
                           root[(size_t)(k0 + 1) * HIDDEN + col]);
    }
}

// ---------------------------------------------------------------------------
// 4) out[m] = x[m] @ root + bias   (WMMA f32 16x16x4; async-DMA x staging)
// ---------------------------------------------------------------------------
__global__ __launch_bounds__(256) void k_root_gemm(
    const float* __restrict__ x, const float2* __restrict__ Pr,
    const float* __restrict__ bias, float* __restrict__ out, int N) {
    __shared__ alignas(16) float xs[16][HIDDEN + 4];   // 132-f rows: 16B aligned + bank-safe

    const int tid   = threadIdx.x;
    const int mtile = blockIdx.x;          // 50000/16 = 3125 exact

    {   // stage 16x128 tile via async global->LDS DMA (2 x b128 per thread)
        const int row = tid & 15;
        const int cb  = (tid >> 4) * 8;
        const float* xp = x + (size_t)(mtile * 16 + row) * HIDDEN + cb;
        const unsigned lo = lds_offset_of(&xs[row][cb]);
        async_copy_b128(lo, xp);
        async_copy_b128(lo + 16, xp + 4);
    }
    async_wait0();
    __syncthreads();

    const int wave = tid >> 5;
    const int lane = tid & 31;
    const int half = lane >> 4;
    const int nloc = lane & 15;
    const int col  = wave * 16 + nloc;
    const int arow = lane & 15;

    const float2* Bp = Pr + (size_t)half * HIDDEN + col;

    v8f acc = {};
    #pragma unroll 4
    for (int ks = 0; ks < 32; ++ks) {
        const int k0 = ks * 4 + half * 2;
        v2f a; a.x = xs[arow][k0];  a.y = xs[arow][k0 + 1];
        float2 bb = Bp[(size_t)ks * 2 * HIDDEN];
        v2f b; b.x = bb.x; b.y = bb.y;
        acc = __builtin_amdgcn_wmma_f32_16x16x4_f32(
            false, a, false, b, (short)0, acc, false, false);
    }

    const float bv = bias[col];
    float* op = out + (size_t)(mtile * 16) * HIDDEN + col;
    #pragma unroll
    for (int j = 0; j < 8; ++j) {
        const int M = j + half * 8;
        op[(size_t)M * HIDDEN] = acc[j] + bv;
    }
}

// ---------------------------------------------------------------------------
// 5) special embedding copy into out[N ..)
// ---------------------------------------------------------------------------
__global__ void k_special(const float* __restrict__ se, float* __restrict__ out, int n) {
    int i = blockIdx.x * blockDim.x + threadIdx.x;
    if (i < n) out[i] = se[i];
}

// ---------------------------------------------------------------------------
// 6) Hot kernel: 8 tiles x 16 edges per block (128 edges).
//    Stacked GEMM: Astack[e, b*128+k] = (comp[rel_e,b]*norm_e) * x[src_e][k]
//    (scale folded into per-row weight -> LDS staging is a raw async DMA copy).
//    B pairs loaded once per (b,ks), reused across the 8 tiles.
// ---------------------------------------------------------------------------
__global__ __launch_bounds__(256) void k_edge_wmma(
    const int* __restrict__ src, const int* __restrict__ dstA,
    const int* __restrict__ et,  const float* __restrict__ x,
    const float2* __restrict__ Pb, const float* __restrict__ comp,
    const unsigned* __restrict__ cnt, float* __restrict__ out, int E, int N) {
    __shared__ alignas(16) float As[TPB][16][HIDDEN + 4];   // 67.5 KB
    __shared__ float wgt[TPB][16][NUM_BASES];   // comp[rel,b] * norm per edge row
    __shared__ int   dstS[TPB][16];
    __shared__ int   srcS[TPB][16];

    const int tid    = threadIdx.x;
    const int base_e = blockIdx.x * EPB;

    // phase 1: per-edge metadata (one thread per edge; pad rows -> wgt 0)
    if (tid < EPB) {
        const int t   = tid >> 4;
        const int row = tid & 15;
        const int e   = base_e + tid;
        int d = -1, s = 0;
        if (e < E) {
            s = src[e];
            d = dstA[e];
            const int r = et[e];
            const unsigned c = cnt[(size_t)r * N + d];
            const float nm = 1.0f / fmaxf((float)c, 1.0f);
            #pragma unroll
            for (int b = 0; b < NUM_BASES; ++b)
                wgt[t][row][b] = nm * comp[r * NUM_BASES + b];
        } else {
            #pragma unroll
            for (int b = 0; b < NUM_BASES; ++b) wgt[t][row][b] = 0.0f;
        }
        dstS[t][row] = d;
        srcS[t][row] = s;
    }
    __syncthreads();

    // phase 2: async DMA gather rows straight into LDS (no VGPR round-trip)
    {
        const int row = tid & 15;
        const int cb  = (tid >> 4) * 8;
        #pragma unroll
        for (int t = 0; t < TPB; ++t) {
            const float* xp = x + (size_t)srcS[t][row] * HIDDEN + cb;
            const unsigned lo = lds_offset_of(&As[t][row][cb]);
            async_copy_b128(lo, xp);
            async_copy_b128(lo + 16, xp + 4);
        }
    }
    async_wait0();
    __syncthreads();

    const int wave = tid >> 5;
    const int lane = tid & 31;
    const int half = lane >> 4;
    const int nloc = lane & 15;
    const int col  = wave * 16 + nloc;
    const int arow = lane & 15;

    v8f acc[TPB] = {};

    for (int b = 0; b < NUM_BASES; ++b) {
        float wt[TPB];
        #pragma unroll
        for (int t = 0; t < TPB; ++t) wt[t] = wgt[t][arow][b];

        const float2* Bb = Pb + ((size_t)(b * 64 + half)) * HIDDEN + col;

        #pragma unroll 4
        for (int ks = 0; ks < 32; ++ks) {
            const int k0 = ks * 4 + half * 2;
            float2 bb = Bb[(size_t)ks * 2 * HIDDEN];
            v2f B; B.x = bb.x; B.y = bb.y;
            #pragma unroll
            for (int t = 0; t < TPB; ++t) {
                v2f a;
                a.x = wt[t] * As[t][arow][k0];
                a.y = wt[t] * As[t][arow][k0 + 1];
                acc[t] = __builtin_amdgcn_wmma_f32_16x16x4_f32(
                    false, a, false, B, (short)0, acc[t], false, false);
            }
        }
    }

    // phase 3: scatter-add into destinations
    #pragma unroll
    for (int t = 0; t < TPB; ++t) {
        #pragma unroll
        for (int j = 0; j < 8; ++j) {
            const int M = j + half * 8;
            const int d = dstS[t][M];
            if (d >= 0) atomicAdd(&out[(size_t)d * HIDDEN + col], acc[t][j]);
        }
    }
}

// ---------------------------------------------------------------------------
extern "C" void kernel_launch(void* const* d_in, const int* in_sizes, int n_in,
                              void* d_out, int out_size, void* d_ws, size_t ws_size,
                              hipStream_t stream) {
    (void)n_in; (void)out_size; (void)ws_size;

    const int*   edge_index = (const int*)d_in[0];   // [2, E]
    const int*   edge_type  = (const int*)d_in[1];   // [E]
    const float* x          = (const float*)d_in[2]; // [N, 128]
    const float* basis      = (const float*)d_in[3]; // [8, 128, 128]
    const float* comp       = (const float*)d_in[4]; // [48, 8]
    const float* root       = (const float*)d_in[5]; // [128, 128]
    const float* bias       = (const float*)d_in[6]; // [128]
    const float* special    = (const float*)d_in[7]; // [S, 128]

    const int E = in_sizes[1];
    const int N = in_sizes[2] / HIDDEN;
    const int S = in_sizes[7] / HIDDEN;

    const int* src = edge_index;        // row 0
    const int* dst = edge_index + E;    // row 1

    float* out = (float*)d_out;

    // workspace layout
    char* ws = (char*)d_ws;
    unsigned* cnt = (unsigned*)ws;                                    // 48*N u32 (9.6 MB)
    float2*   Pb  = (float2*)(ws + (size_t)NUM_REL * N * 4);          // 512 KB
    float2*   Pr  = (float2*)(ws + (size_t)NUM_REL * N * 4 + 65536 * sizeof(float2)); // 64 KB

    const int cntN = NUM_REL * N;
    k_zero_u32<<<(cntN + 255) / 256, 256, 0, stream>>>(cnt, cntN);
    k_hist<<<(E + 255) / 256, 256, 0, stream>>>(edge_type, dst, cnt, E, N);

    k_pack_basis<<<(NUM_BASES * 32 * 2 * HIDDEN + 255) / 256, 256, 0, stream>>>(basis, Pb);
    k_pack_root<<<(32 * 2 * HIDDEN + 255) / 256, 256, 0, stream>>>(root, Pr);

    k_root_gemm<<<N / 16, 256, 0, stream>>>(x, Pr, bias, out, N);
    k_special<<<(S * HIDDEN + 255) / 256, 256, 0, stream>>>(
        special, out + (size_t)N * HIDDEN, S * HIDDEN);

    k_edge_wmma<<<(E + EPB - 1) / EPB, 256, 0, stream>>>(
        src, dst, edge_type, x, Pb, comp, cnt, out, E, N);
}